// NeuralODETayMode_62156766708276
// MI455X (gfx1250) — compile-verified
//
#include <hip/hip_runtime.h>
#include <cmath>

// ---------------------------------------------------------------------------
// Neural ODE (Taylor-mode regularized) on MI455X / gfx1250.
// f32 end-to-end; GEMVs use V_WMMA_F32_16X16X4_F32 (f32-exact matrix pipe),
// 4-way K-split per 16-column group + dual accumulators (2x unroll) to break
// the WMMA dependency chain. Input staging (concat/sigmoid/RK-axpy/tangent
// seed) fused into gemv's LDS staging phase. Fixed-step RK4 (64 steps)
// replaces adaptive Dopri5 deterministically.
// ---------------------------------------------------------------------------

typedef __attribute__((ext_vector_type(2))) float v2f;
typedef __attribute__((ext_vector_type(8))) float v8f;

#define STATE  4096
#define CDIM   1024
#define FANIN  5121    // CDIM + STATE + 1
#define NSTEPS 64

__device__ __forceinline__ float sigm(float v) { return 1.0f / (1.0f + expf(-v)); }

// ---- state init: y <- h, r <- 0 -------------------------------------------
__global__ void init_kernel(const float* __restrict__ h,
                            float* __restrict__ y, float* __restrict__ r, int n) {
  int i = blockIdx.x * blockDim.x + threadIdx.x;
  if (i < n) y[i] = h[i];
  if (i == 0) r[0] = 0.0f;
}

// ---- fused staging + GEMV via V_WMMA_F32_16X16X4_F32 ----------------------
// Staging into LDS (per block), with yeff[j] = src[j] + coefK*h*srcK[j]:
//   mode 0: xs[j] = src[j]                         (xlen = STATE)
//   mode 1: xs = [c, sigmoid(yeff), t*tfrac]       (xlen = FANIN)
//   mode 2: xs = [c, src, t*tfrac]                 (xlen = FANIN, src pre-activated)
//   mode 3: xs[j] = sig'(yeff[j]) * src2[j]        (xlen = STATE)
// Compute: out[n] = post( sum_{k<4*K4} xs[k]*W[krow0+k][n]
//                         + tail*W[krow0+4*K4][n] + b[n] )
//   tail = tail_scale_from_x ? xs[4*K4] : 1.0  (time slot / dt=1 tangent slot)
//   post: optional sigmoid, or multiply by s*(1-s) from sigmul buffer.
// WMMA mapping (16x16x4 f32): A = x chunk replicated over all 16 M rows
//   (vgpr0 = xs[k+2*half], vgpr1 = xs[k+2*half+1], half = lane>=16);
//   B = W[k+2*half+{0,1}][col0+lane%16]; D rows identical -> acc[0] lanes 0-15.
// Block: 8 waves = 2 column-groups x 4 K-splits; dual accumulators per wave;
// partials reduced via LDS.
__global__ void __launch_bounds__(256)
gemv_wmma_kernel(const float* __restrict__ src, const float* __restrict__ srcK, float coefK,
                 const float* __restrict__ src2, const float* __restrict__ c,
                 const float* __restrict__ tptr, float tfrac, float inv_nsteps,
                 int stage_mode, int xlen, int K4, int krow0,
                 const float* __restrict__ W, int N,
                 const float* __restrict__ bias, const float* __restrict__ sigmul,
                 int tail_scale_from_x, int act_sigmoid,
                 float* __restrict__ out) {
  __shared__ float xs[FANIN];
  __shared__ float partial[8][16];
  const int tid = threadIdx.x;
  const float hstep = tptr[0] * inv_nsteps;

  for (int i = tid; i < xlen; i += 256) {
    float v;
    if (stage_mode == 0) {
      v = src[i];
    } else if (stage_mode <= 2) {
      if (i < CDIM) {
        v = c[i];
      } else if (i < CDIM + STATE) {
        float y = src[i - CDIM];
        if (srcK) y += coefK * hstep * srcK[i - CDIM];
        v = (stage_mode == 1) ? sigm(y) : y;
      } else {
        v = tptr[0] * tfrac;
      }
    } else {  // mode 3: tangent seed
      float y = src[i];
      if (srcK) y += coefK * hstep * srcK[i];
      float s = sigm(y);
      v = s * (1.0f - s) * src2[i];
    }
    xs[i] = v;
  }
  __syncthreads();

  const int lane   = tid & 31;
  const int wave   = tid >> 5;
  const int colgrp = wave >> 2;   // 0..1
  const int ksplit = wave & 3;    // 0..3
  const int half   = lane >> 4;   // A/B rows: 0 -> K=0/1, 1 -> K=2/3
  const int l16    = lane & 15;
  const int col0   = (blockIdx.x * 2 + colgrp) * 16;
  const int Kq     = K4 >> 2;     // chunks per K-split (1280->320, 1024->256); even
  const int kc0    = ksplit * Kq;
  const int krows  = 4 * K4;

  v8f acc0 = {};
  v8f acc1 = {};
  const float* wbase = W + (size_t)krow0 * N + col0 + l16;
  for (int kc = kc0; kc < kc0 + Kq; kc += 2) {
    const int k = 4 * kc;
    v2f A0, B0, A1, B1;
    A0.x = xs[k + 2 * half];
    A0.y = xs[k + 2 * half + 1];
    A1.x = xs[k + 4 + 2 * half];
    A1.y = xs[k + 4 + 2 * half + 1];
    const float* wp0 = wbase + (size_t)(k + 2 * half) * N;
    const float* wp1 = wbase + (size_t)(k + 4 + 2 * half) * N;
    B0.x = wp0[0];
    B0.y = wp0[N];
    B1.x = wp1[0];
    B1.y = wp1[N];
    // stream-ahead prefetch (64 rows ahead, clamped): global_prefetch_b8
    int kpf = k + 64; if (kpf > krows - 4) kpf = krows - 4;
    __builtin_prefetch(wbase + (size_t)(kpf + 2 * half) * N, 0, 0);
    // Two independent accumulator chains -> back-to-back WMMA issue.
    acc0 = __builtin_amdgcn_wmma_f32_16x16x4_f32(
        /*neg_a=*/false, A0, /*neg_b=*/false, B0,
        /*c_mod=*/(short)0, acc0, /*reuse_a=*/false, /*reuse_b=*/false);
    acc1 = __builtin_amdgcn_wmma_f32_16x16x4_f32(
        /*neg_a=*/false, A1, /*neg_b=*/false, B1,
        /*c_mod=*/(short)0, acc1, /*reuse_a=*/false, /*reuse_b=*/false);
  }
  if (lane < 16) partial[wave][l16] = acc0[0] + acc1[0];
  __syncthreads();

  if (tid < 32) {
    const int cg   = tid >> 4;
    const int cl   = tid & 15;
    const int colw = (blockIdx.x * 2 + cg) * 16 + cl;
    float v = partial[cg * 4 + 0][cl] + partial[cg * 4 + 1][cl]
            + partial[cg * 4 + 2][cl] + partial[cg * 4 + 3][cl];
    const float tscale = tail_scale_from_x ? xs[krows] : 1.0f;
    v += tscale * W[(size_t)(krow0 + krows) * N + colw];
    if (bias) v += bias[colw];
    if (act_sigmoid) v = sigm(v);
    if (sigmul) { float s = sigmul[colw]; v *= s * (1.0f - s); }
    out[colw] = v;
  }
}

// ---- dr = mean(g^2) --------------------------------------------------------
__global__ void meansq_kernel(const float* __restrict__ g, float* __restrict__ outv, int n) {
  __shared__ float red[256];
  float s = 0.0f;
  for (int i = threadIdx.x; i < n; i += 256) { float v = g[i]; s += v * v; }
  red[threadIdx.x] = s;
  __syncthreads();
  for (int off = 128; off > 0; off >>= 1) {
    if (threadIdx.x < off) red[threadIdx.x] += red[threadIdx.x + off];
    __syncthreads();
  }
  if (threadIdx.x == 0) outv[0] = red[0] / (float)n;
}

// ---- RK4 combine for y and r ----------------------------------------------
__global__ void rk4_update_kernel(float* __restrict__ y,
                                  const float* __restrict__ k1, const float* __restrict__ k2,
                                  const float* __restrict__ k3, const float* __restrict__ k4,
                                  float* __restrict__ r,
                                  const float* __restrict__ d1, const float* __restrict__ d2,
                                  const float* __restrict__ d3, const float* __restrict__ d4,
                                  const float* __restrict__ tptr, float inv_nsteps, int n) {
  int i = blockIdx.x * blockDim.x + threadIdx.x;
  float h6 = tptr[0] * inv_nsteps * (1.0f / 6.0f);
  if (i < n) y[i] += h6 * (k1[i] + 2.0f * k2[i] + 2.0f * k3[i] + k4[i]);
  if (i == 0) r[0] += h6 * (d1[0] + 2.0f * d2[0] + 2.0f * d3[0] + d4[0]);
}

// ---- pack outputs ----------------------------------------------------------
__global__ void finalize_kernel(const float* __restrict__ y, const float* __restrict__ r,
                                float* __restrict__ outv, int n) {
  int i = blockIdx.x * blockDim.x + threadIdx.x;
  if (i < n) outv[i] = y[i];
  if (i == n) outv[n] = r[0];
}

extern "C" void kernel_launch(void* const* d_in, const int* in_sizes, int n_in,
                              void* d_out, int out_size, void* d_ws, size_t ws_size,
                              hipStream_t stream) {
  const float* h_in = (const float*)d_in[0];
  const float* tptr = (const float*)d_in[1];
  const float* c    = (const float*)d_in[2];
  const float* W1   = (const float*)d_in[3];
  const float* b1   = (const float*)d_in[4];
  const float* W2   = (const float*)d_in[5];
  const float* b2   = (const float*)d_in[6];

  float* ws = (float*)d_ws;
  // workspace layout (floats); total ~131 KB
  float* Y  = ws + 0;       // state y            [4096]
  float* R  = ws + 4096;    // state r            [1]
  float* S1 = ws + 4160;    // sigmoid(x@W1+b1)   [4096]
  float* V1 = ws + 8256;    // s1'(u)*udot        [4096]
  float* GD = ws + 12352;   // g = d/dt f         [4096]
  float* K0 = ws + 16448;   // RK stages k1..k4   [4x4096]
  float* K1 = ws + 20544;
  float* K2 = ws + 24640;
  float* K3 = ws + 28736;
  float* DR = ws + 32832;   // stage dr scalars   [4]

  const float INV = 1.0f / (float)NSTEPS;
  const int GB = 16;   // blocks for 4096-wide elementwise
  const int MB = 128;  // gemv blocks: 128 blk * 2 colgrp * 16 cols = 4096

  // Augmented dynamics at yeff = Y + coefK*h*srcK, t = tptr[0]*tfrac
  //   -> kout[4096] = f(yeff,t), drout[1] = mean((J_f.f + df/dt)^2)
  auto eval = [&](const float* srcK, float coefK, float tfrac, float* kout, float* drout) {
    // s1 = sigmoid([c, sigmoid(yeff), t] @ W1 + b1)
    gemv_wmma_kernel<<<MB, 256, 0, stream>>>(Y, srcK, coefK, nullptr, c, tptr, tfrac, INV,
                                             /*mode*/1, FANIN, 1280, 0, W1, STATE,
                                             b1, nullptr, /*tail=t*/1, /*sig*/1, S1);
    // f = [c, s1, t] @ W2 + b2
    gemv_wmma_kernel<<<MB, 256, 0, stream>>>(S1, nullptr, 0.0f, nullptr, c, tptr, tfrac, INV,
                                             /*mode*/2, FANIN, 1280, 0, W2, STATE,
                                             b2, nullptr, 1, 0, kout);
    // v1 = s1'(u) * ( sig'(yeff)*f @ W1[CDIM:,:] + W1[last] )
    gemv_wmma_kernel<<<MB, 256, 0, stream>>>(Y, srcK, coefK, kout, nullptr, tptr, tfrac, INV,
                                             /*mode*/3, STATE, 1024, CDIM, W1, STATE,
                                             nullptr, S1, /*tail=1.0*/0, 0, V1);
    // g = v1 @ W2[CDIM:,:] + W2[last]
    gemv_wmma_kernel<<<MB, 256, 0, stream>>>(V1, nullptr, 0.0f, nullptr, nullptr, tptr, tfrac, INV,
                                             /*mode*/0, STATE, 1024, CDIM, W2, STATE,
                                             nullptr, nullptr, 0, 0, GD);
    meansq_kernel<<<1, 256, 0, stream>>>(GD, drout, STATE);
  };

  init_kernel<<<GB, 256, 0, stream>>>(h_in, Y, R, STATE);

  for (int s = 0; s < NSTEPS; ++s) {
    const float fs = (float)s;
    eval(nullptr, 0.0f, fs * INV,          K0, DR + 0);
    eval(K0,      0.5f, (fs + 0.5f) * INV, K1, DR + 1);
    eval(K1,      0.5f, (fs + 0.5f) * INV, K2, DR + 2);
    eval(K2,      1.0f, (fs + 1.0f) * INV, K3, DR + 3);
    rk4_update_kernel<<<GB, 256, 0, stream>>>(Y, K0, K1, K2, K3, R,
                                              DR + 0, DR + 1, DR + 2, DR + 3,
                                              tptr, INV, STATE);
  }

  finalize_kernel<<<17, 256, 0, stream>>>(Y, R, (float*)d_out, STATE);
}